// SparseTokenHandler_59631325937850
// MI455X (gfx1250) — compile-verified
//
#include <hip/hip_runtime.h>
#include <hip/hip_bf16.h>

typedef __attribute__((ext_vector_type(2))) float v2f;
typedef __attribute__((ext_vector_type(8))) float v8f;

#define B_ 4
#define L_ 16384
#define C_ 512
// keep_len = L/2 = 8192 ; num_random = int(8192*0.1) = 819 ; num_top = 8192-819
#define NUM_TOP 7373
#define NUM_RAND 819

// ---------------------------------------------------------------------------
// Kernel 1: per-token L2-norm^2 scores via V_WMMA_F32_16X16X4_F32.
// Row-sum trick: A[i,k] = x[i,c0+k]^2, B = ones(4x16)  =>  D[i,j] = partial score of token i.
// f32 16x4 A layout: lanes 0-15 hold M=0..15 with K={0,1} in v0/v1; lanes 16-31 hold K={2,3}.
// C/D layout: VGPR r, lanes 0-15 -> M=r; lanes 16-31 -> M=8+r. All columns equal, so
// lane 0 holds scores of tokens 0..7 in acc[0..7], lane 16 holds tokens 8..15.
// ---------------------------------------------------------------------------
__global__ void score_kernel(const float* __restrict__ x, float* __restrict__ scores) {
  const int lane = threadIdx.x & 31;
  const int waveInBlock = threadIdx.x >> 5;
  const int wid = blockIdx.x * (blockDim.x >> 5) + waveInBlock;
  const int tokBase = wid * 16;
  const int tok = tokBase + (lane & 15);
  const int half = lane >> 4;

  const float2* row = (const float2*)(x + (size_t)tok * C_);
  v8f acc = {};
  v2f ones;
  ones[0] = 1.0f;
  ones[1] = 1.0f;

  #pragma unroll 4
  for (int k0 = 0; k0 < C_; k0 += 4) {
    float2 a = row[(k0 >> 1) + half];
    v2f av;
    av[0] = a.x * a.x;
    av[1] = a.y * a.y;
    // 8 args: (neg_a, A, neg_b, B, c_mod, C, reuse_a, reuse_b)
    acc = __builtin_amdgcn_wmma_f32_16x16x4_f32(
        false, av, false, ones, (short)0, acc, false, false);
  }

  if (lane == 0) {
    #pragma unroll
    for (int r = 0; r < 8; ++r) scores[tokBase + r] = acc[r];
  } else if (lane == 16) {
    #pragma unroll
    for (int r = 0; r < 8; ++r) scores[tokBase + 8 + r] = acc[r];
  }
}

// ---------------------------------------------------------------------------
// Kernel 2: exact top-k selection per batch (one workgroup per batch).
// 64-bit key = (monotone float bits << 32) | (L-1-i)  -> total order matching
// jax.lax.top_k (descending value, ties broken by lower index).
// MSB radix-select (8 passes x 8 bits) with an LDS histogram finds the k-th
// largest key exactly; keep iff key >= threshold (keys are unique).
// ---------------------------------------------------------------------------
__device__ __forceinline__ unsigned monotone_bits(float f) {
  unsigned u = __float_as_uint(f);
  return u ^ ((u & 0x80000000u) ? 0xFFFFFFFFu : 0x80000000u);
}

__device__ unsigned long long radix_select_desc(const unsigned long long* __restrict__ k,
                                                int kwant, unsigned* hist,
                                                unsigned long long* s_prefix, int* s_krem) {
  const int tid = threadIdx.x;
  const int nthr = blockDim.x;
  if (tid == 0) { *s_prefix = 0ull; *s_krem = kwant; }
  __syncthreads();
  for (int shift = 56; shift >= 0; shift -= 8) {
    for (int j = tid; j < 256; j += nthr) hist[j] = 0u;
    __syncthreads();
    const unsigned long long prefMask = (shift == 56) ? 0ull : (~0ull << (shift + 8));
    const unsigned long long pref = *s_prefix;
    for (int i = tid; i < L_; i += nthr) {
      unsigned long long key = k[i];
      if ((key & prefMask) == pref)
        atomicAdd(&hist[(int)((key >> shift) & 0xFF)], 1u);
    }
    __syncthreads();
    if (tid == 0) {
      int kk = *s_krem;
      unsigned cum = 0;
      for (int d = 255; d >= 0; --d) {
        unsigned c = hist[d];
        if (cum + c >= (unsigned)kk) {
          *s_prefix = pref | ((unsigned long long)d << shift);
          *s_krem = kk - (int)cum;
          break;
        }
        cum += c;
      }
    }
    __syncthreads();
  }
  return *s_prefix;
}

__global__ void select_kernel(const float* __restrict__ scores,
                              const float* __restrict__ noise,
                              unsigned long long* __restrict__ keysGlobal,
                              unsigned* __restrict__ keepGlobal) {
  const int b = blockIdx.x;
  const int tid = threadIdx.x;
  const int nthr = blockDim.x;
  unsigned long long* k = keysGlobal + (size_t)b * L_;
  unsigned* kp = keepGlobal + (size_t)b * L_;

  __shared__ unsigned hist[256];
  __shared__ unsigned long long s_prefix;
  __shared__ int s_krem;

  // ---- phase A: top-NUM_TOP by score ----
  for (int i = tid; i < L_; i += nthr) {
    unsigned m = monotone_bits(scores[(size_t)b * L_ + i]);
    k[i] = ((unsigned long long)m << 32) | (unsigned)(L_ - 1 - i);
  }
  __threadfence();
  __syncthreads();
  unsigned long long thrTop = radix_select_desc(k, NUM_TOP, hist, &s_prefix, &s_krem);
  for (int i = tid; i < L_; i += nthr) kp[i] = (k[i] >= thrTop) ? 1u : 0u;
  __threadfence();
  __syncthreads();

  // ---- phase B: top-NUM_RAND by noise among non-top tokens ----
  for (int i = tid; i < L_; i += nthr) {
    unsigned long long key = 0ull;  // masked (already kept) -> smallest possible key
    if (!kp[i]) {
      unsigned m = monotone_bits(noise[(size_t)b * L_ + i]);
      key = ((unsigned long long)m << 32) | (unsigned)(L_ - 1 - i);
    }
    k[i] = key;
  }
  __threadfence();
  __syncthreads();
  unsigned long long thrRnd = radix_select_desc(k, NUM_RAND, hist, &s_prefix, &s_krem);
  for (int i = tid; i < L_; i += nthr)
    if (!kp[i] && k[i] != 0ull && k[i] >= thrRnd) kp[i] = 1u;
}

// ---------------------------------------------------------------------------
// Kernel 3: densify. One block per token; 128 threads x 16B = 2KB row.
// Kept rows move via the CDNA5 async data movers: global -> LDS -> global,
// never transiting VGPRs (tracked with ASYNCcnt). Non-kept rows stream zeros.
// ---------------------------------------------------------------------------
__global__ void densify_kernel(const float* __restrict__ x,
                               const unsigned* __restrict__ keep,
                               float* __restrict__ out) {
  __shared__ __align__(16) float4 buf[128];
  const int tok = blockIdx.x;
  const size_t base = (size_t)tok * C_ + (size_t)threadIdx.x * 4;
  if (keep[tok]) {
    const float* src = x + base;
    float* dst = out + base;
    unsigned lds = (unsigned)(unsigned long long)&buf[threadIdx.x];
    asm volatile("global_load_async_to_lds_b128 %0, %1, off"
                 :: "v"(lds), "v"(src) : "memory");
    asm volatile("s_wait_asynccnt 0x0" ::: "memory");
    asm volatile("global_store_async_from_lds_b128 %0, %1, off"
                 :: "v"(dst), "v"(lds) : "memory");
    asm volatile("s_wait_asynccnt 0x0" ::: "memory");
  } else {
    *(float4*)(out + base) = make_float4(0.f, 0.f, 0.f, 0.f);
  }
}

extern "C" void kernel_launch(void* const* d_in, const int* in_sizes, int n_in,
                              void* d_out, int out_size, void* d_ws, size_t ws_size,
                              hipStream_t stream) {
  const float* x = (const float*)d_in[0];
  const float* noise = (const float*)d_in[1];
  float* out = (float*)d_out;

  char* ws = (char*)d_ws;
  float* scores = (float*)ws;                                                    // 256 KB
  unsigned long long* keys = (unsigned long long*)(ws + (size_t)B_ * L_ * 4);    // 512 KB
  unsigned* keep = (unsigned*)(ws + (size_t)B_ * L_ * 4 + (size_t)B_ * L_ * 8);  // 256 KB

  // 1 wave -> 16 tokens; 8 waves (256 thr) per block -> 128 tokens/block
  const int blocks1 = (B_ * L_) / 128;  // 512
  score_kernel<<<blocks1, 256, 0, stream>>>(x, scores);
  select_kernel<<<B_, 256, 0, stream>>>(scores, noise, keys, keep);
  densify_kernel<<<B_ * L_, 128, 0, stream>>>(x, keep, out);
}